// LSTM_62543313764830
// MI455X (gfx1250) — compile-verified
//
#include <hip/hip_runtime.h>

typedef __attribute__((ext_vector_type(2))) float v2f;
typedef __attribute__((ext_vector_type(8))) float v8f;

#define BB 4096
#define SS 512
#define FF 8
#define HH 10

// ---- activations on the CDNA5 TRANS pipe ----
__device__ __forceinline__ float fast_tanh(float x) {
#if __has_builtin(__builtin_amdgcn_tanhf)
  return __builtin_amdgcn_tanhf(x);
#else
  x = fminf(fmaxf(x, -15.0f), 15.0f);
  float e = __expf(2.0f * x);
  return (e - 1.0f) * __builtin_amdgcn_rcpf(e + 1.0f);
#endif
}
__device__ __forceinline__ float fast_sig(float x) {
  return fmaf(0.5f, fast_tanh(0.5f * x), 0.5f);
}

__device__ __forceinline__ v8f wmma4(v2f a, v2f b, v8f c) {
  // D = A(16x4 f32) * B(4x16 f32) + C(16x16 f32), full fp32 precision
  return __builtin_amdgcn_wmma_f32_16x16x4_f32(false, a, false, b, (short)0, c, false, false);
}

__global__ __launch_bounds__(32)
void lstm2_wmma_kernel(const float* __restrict__ X,
                       const float* __restrict__ Hin,
                       const float* __restrict__ Cin,
                       const float* __restrict__ Wih0, const float* __restrict__ Whh0,
                       const float* __restrict__ bih0, const float* __restrict__ bhh0,
                       const float* __restrict__ Wih1, const float* __restrict__ Whh1,
                       const float* __restrict__ bih1, const float* __restrict__ bhh1,
                       const float* __restrict__ Wfc,  const float* __restrict__ bfc,
                       float* __restrict__ Out)
{
  // Rows 0..15: [h_l0(10) | h_l1(10)] packed, stride 20 (floats 0..319).
  // Floats 336..479: dump region for lanes with nl>=10 (unconditional stores).
  __shared__ float buf[480];

  const int lane = threadIdx.x & 31;
  const int nl   = lane & 15;   // column within N-tile / row for A-frag
  const int half = lane >> 4;   // K-pair selector within fragment
  const int tb   = blockIdx.x * 16;
  const bool cval = (nl < HH);

  // Unconditional-store bases: valid lanes hit the packed row, junk lanes hit dump.
  const int sb0 = cval ? (8 * half * 20 + nl)      : 336;
  const int sb1 = cval ? (8 * half * 20 + nl + HH) : 336;

  // ---- preload weight B-fragments: bf[chunk][gate-tile][vgpr] ----
  float bf0[5][4][2];
  float bf1[5][4][2];
#pragma unroll
  for (int c = 0; c < 5; ++c) {
#pragma unroll
    for (int t = 0; t < 4; ++t) {
      const int gr = t * HH + nl;
#pragma unroll
      for (int v = 0; v < 2; ++v) {
        const int k = 4 * c + 2 * half + v;
        float w0 = 0.f, w1 = 0.f;
        if (cval) {
          if (k < FF)            w0 = Wih0[gr * FF + k];
          else if (k < FF + HH)  w0 = Whh0[gr * HH + (k - FF)];
          if (k < HH)            w1 = Wih1[gr * HH + k];
          else                   w1 = Whh1[gr * HH + (k - HH)];  // k<20 always
        }
        bf0[c][t][v] = w0;
        bf1[c][t][v] = w1;
      }
    }
  }

  // biases broadcast into C-init fragments (bias depends only on column)
  v8f bias0[4], bias1[4];
#pragma unroll
  for (int t = 0; t < 4; ++t) {
    float b0 = cval ? (bih0[t * HH + nl] + bhh0[t * HH + nl]) : 0.f;
    float b1 = cval ? (bih1[t * HH + nl] + bhh1[t * HH + nl]) : 0.f;
    v8f v0, v1;
#pragma unroll
    for (int r = 0; r < 8; ++r) { v0[r] = b0; v1[r] = b1; }
    bias0[t] = v0; bias1[t] = v1;
  }

  // ---- initial state ----
  v8f c0, c1;  // cell state, C-layout registers (never needs relayout)
#pragma unroll
  for (int r = 0; r < 8; ++r) {
    int row = r + 8 * half;
    c0[r] = cval ? Cin[(size_t)(tb + row) * HH + nl] : 0.f;
    c1[r] = cval ? Cin[(size_t)BB * HH + (size_t)(tb + row) * HH + nl] : 0.f;
  }
  for (int idx = lane; idx < 16 * 20; idx += 32) {
    int row = idx / 20, col = idx % 20;
    buf[idx] = (col < HH) ? Hin[(size_t)(tb + row) * HH + col]
                          : Hin[(size_t)BB * HH + (size_t)(tb + row) * HH + (col - HH)];
  }
  __syncthreads();

  const float* xrow = X + ((size_t)(tb + nl) * SS) * FF;
  const float* hr   = &buf[nl * 20];
  const int    k0   = 2 * half;

  // ---- software-pipelined fetches ----
  // x for step 0:
  v2f nx0, nx1;
  nx0.x = xrow[0 + k0]; nx0.y = xrow[1 + k0];
  nx1.x = xrow[4 + k0]; nx1.y = xrow[5 + k0];
  // shared h chunks (buf cols 0..11): serve layer-0 recurrent input of this step
  // AND layer-1 chunks 0..2 of the previous step. Preload for step 0.
  v2f hc0, hc1, hc2;
  hc0.x = hr[0 + k0]; hc0.y = hr[1 + k0];
  hc1.x = hr[4 + k0]; hc1.y = hr[5 + k0];
  hc2.x = hr[8 + k0]; hc2.y = hr[9 + k0];

  for (int s = 0; s < SS; ++s) {
    // consume prefetched x, issue next step's global loads immediately
    v2f ax0 = nx0, ax1 = nx1;
    {
      int sn = s + 1; if (sn >= SS) sn = SS - 1;  // clamp: no OOB read
      const float* xp = xrow + (size_t)sn * FF;
      nx0.x = xp[0 + k0]; nx0.y = xp[1 + k0];
      nx1.x = xp[4 + k0]; nx1.y = xp[5 + k0];
    }
    // layer-1 chunks 3,4 (buf cols 12..19) depend only on PREVIOUS step's h_l1:
    // issue early so their latency hides under the layer-0 WMMA/tanh chain.
    v2f a3, a4;
    a3.x = hr[12 + k0]; a3.y = hr[13 + k0];
    a4.x = hr[16 + k0]; a4.y = hr[17 + k0];

    // ================= layer 0 ================= (A: ax0,ax1,hc0,hc1,hc2)
    v8f acc0[4];
#pragma unroll
    for (int t = 0; t < 4; ++t) {
      v2f b; v8f a = bias0[t];
      b.x = bf0[0][t][0]; b.y = bf0[0][t][1]; a = wmma4(ax0, b, a);
      b.x = bf0[1][t][0]; b.y = bf0[1][t][1]; a = wmma4(ax1, b, a);
      b.x = bf0[2][t][0]; b.y = bf0[2][t][1]; a = wmma4(hc0, b, a);
      b.x = bf0[3][t][0]; b.y = bf0[3][t][1]; a = wmma4(hc1, b, a);
      b.x = bf0[4][t][0]; b.y = bf0[4][t][1]; a = wmma4(hc2, b, a);
      acc0[t] = a;
    }
#pragma unroll
    for (int r = 0; r < 8; ++r) {
      float iv = fast_sig(acc0[0][r]);
      float fv = fast_sig(acc0[1][r]);
      float gv = fast_tanh(acc0[2][r]);
      float ov = fast_sig(acc0[3][r]);
      float cn = fmaf(fv, c0[r], iv * gv);
      c0[r] = cn;
      float hn = ov * fast_tanh(cn);
      buf[sb0 + r * 20] = hn;   // unconditional: junk lanes hit dump region
    }

    // Reload shared chunks (cols 0..11): fresh h_l0 just stored; cols 10,11 are
    // previous-step h_l1 (correct for layer-1 K=10,11; multiplied by zero
    // weights when reused as layer-0 K=18,19 next step).
    hc0.x = hr[0 + k0]; hc0.y = hr[1 + k0];
    hc1.x = hr[4 + k0]; hc1.y = hr[5 + k0];
    hc2.x = hr[8 + k0]; hc2.y = hr[9 + k0];

    // ================= layer 1 ================= (A: hc0,hc1,hc2,a3,a4)
    v8f acc1[4];
#pragma unroll
    for (int t = 0; t < 4; ++t) {
      v2f b; v8f a = bias1[t];
      b.x = bf1[0][t][0]; b.y = bf1[0][t][1]; a = wmma4(hc0, b, a);
      b.x = bf1[1][t][0]; b.y = bf1[1][t][1]; a = wmma4(hc1, b, a);
      b.x = bf1[2][t][0]; b.y = bf1[2][t][1]; a = wmma4(hc2, b, a);
      b.x = bf1[3][t][0]; b.y = bf1[3][t][1]; a = wmma4(a3, b, a);
      b.x = bf1[4][t][0]; b.y = bf1[4][t][1]; a = wmma4(a4, b, a);
      acc1[t] = a;
    }
#pragma unroll
    for (int r = 0; r < 8; ++r) {
      float iv = fast_sig(acc1[0][r]);
      float fv = fast_sig(acc1[1][r]);
      float gv = fast_tanh(acc1[2][r]);
      float ov = fast_sig(acc1[3][r]);
      float cn = fmaf(fv, c1[r], iv * gv);
      c1[r] = cn;
      float hn = ov * fast_tanh(cn);
      buf[sb1 + r * 20] = hn;   // unconditional
    }
    // no barrier needed: single wave, DS pipe is in-order per wave
  }

  // ================= outputs =================
  // d_out = [out_regression (B) | hidden_out (2*B*H) | cell_out (2*B*H)]
  const size_t base_h = (size_t)BB;
  const size_t base_c = (size_t)BB + 2ull * BB * HH;
  if (half == 0) {
    const int b = tb + nl;
    float acc = bfc[0];
#pragma unroll
    for (int j = 0; j < HH; ++j) acc += buf[nl * 20 + HH + j] * Wfc[j];
    Out[b] = acc;
#pragma unroll
    for (int j = 0; j < HH; ++j) {
      Out[base_h + (size_t)b * HH + j]                     = buf[nl * 20 + j];
      Out[base_h + (size_t)BB * HH + (size_t)b * HH + j]   = buf[nl * 20 + HH + j];
    }
  }
#pragma unroll
  for (int r = 0; r < 8; ++r) {
    int row = r + 8 * half;
    if (cval) {
      Out[base_c + (size_t)(tb + row) * HH + nl]                   = c0[r];
      Out[base_c + (size_t)BB * HH + (size_t)(tb + row) * HH + nl] = c1[r];
    }
  }
}

extern "C" void kernel_launch(void* const* d_in, const int* in_sizes, int n_in,
                              void* d_out, int out_size, void* d_ws, size_t ws_size,
                              hipStream_t stream) {
  const float* X    = (const float*)d_in[0];
  const float* Hin  = (const float*)d_in[1];
  const float* Cin  = (const float*)d_in[2];
  const float* Wih0 = (const float*)d_in[3];
  const float* Whh0 = (const float*)d_in[4];
  const float* bih0 = (const float*)d_in[5];
  const float* bhh0 = (const float*)d_in[6];
  const float* Wih1 = (const float*)d_in[7];
  const float* Whh1 = (const float*)d_in[8];
  const float* bih1 = (const float*)d_in[9];
  const float* bhh1 = (const float*)d_in[10];
  const float* Wfc  = (const float*)d_in[11];
  const float* bfc  = (const float*)d_in[12];
  float* Out = (float*)d_out;

  dim3 grid(BB / 16), block(32);
  hipLaunchKernelGGL(lstm2_wmma_kernel, grid, block, 0, stream,
                     X, Hin, Cin, Wih0, Whh0, bih0, bhh0,
                     Wih1, Whh1, bih1, bhh1, Wfc, bfc, Out);
}